// OffsetPostModel_60309930770647
// MI455X (gfx1250) — compile-verified
//
#include <hip/hip_runtime.h>
#include <hip/hip_bf16.h>
#include <stdint.h>

// ---------------------------------------------------------------------------
// OffsetPostModel post-processing for MI455X (gfx1250, wave32).
//   K1: 3x3 NMS over (1080,1920,2) heatmap, collect peaks with score>0.5 as
//       u64 keys (value_bits<<32 | ~index) via atomic append.
//       Heatmap tiles staged into LDS with gfx1250 async global->LDS loads.
//   K2: per-channel exact top-15 by (value desc, index asc): 15 argmax passes.
//   K3: gather size/offset maps at winners, emit (15,16) f32 output.
// ---------------------------------------------------------------------------

#define IMG_H 1080
#define IMG_W 1920
#define TW 64
#define TH 4
#define N_OBJS 15
typedef unsigned long long u64;
typedef int v2i __attribute__((ext_vector_type(2)));
typedef __attribute__((address_space(1))) v2i* g_v2i_ptr;
typedef __attribute__((address_space(3))) v2i* l_v2i_ptr;

#if defined(__has_builtin)
#if __has_builtin(__builtin_amdgcn_global_load_async_to_lds_b64)
#define USE_ASYNC_LDS 1
#endif
#endif

__global__ void init_ws_kernel(unsigned* cnt) {
  if (threadIdx.x < 4) cnt[threadIdx.x] = 0u;  // cnt[0..1], nsel[0..1]
}

// ---- Kernel 1: NMS + candidate collection --------------------------------
__global__ __launch_bounds__(256)
void nms_collect_kernel(const float* __restrict__ hm, u64* __restrict__ cand,
                        unsigned* __restrict__ cnt, unsigned cap) {
  __shared__ float2 tile[TH + 2][TW + 2];  // both channels as float2, +halo

  const int tid = threadIdx.x;
  const int x0 = blockIdx.x * TW;
  const int y0 = blockIdx.y * TH;

  // Stage (TH+2)x(TW+2) clamped-neighborhood pixels into LDS.
  // Clamped coords are exactly equivalent to SAME pooling with -inf pad.
  for (int l = tid; l < (TH + 2) * (TW + 2); l += 256) {
    const int lyl = l / (TW + 2);
    const int lxl = l % (TW + 2);
    int gy = y0 + lyl - 1;
    int gx = x0 + lxl - 1;
    gy = gy < 0 ? 0 : (gy > IMG_H - 1 ? IMG_H - 1 : gy);
    gx = gx < 0 ? 0 : (gx > IMG_W - 1 ? IMG_W - 1 : gx);
    const float* src = hm + 2u * ((unsigned)gy * IMG_W + (unsigned)gx);
#ifdef USE_ASYNC_LDS
    __builtin_amdgcn_global_load_async_to_lds_b64(
        (g_v2i_ptr)src, (l_v2i_ptr)&tile[lyl][lxl],
        /*offset=*/0, /*cpol=*/0);
#else
    tile[lyl][lxl] = *(const float2*)src;
#endif
  }
#ifdef USE_ASYNC_LDS
#if __has_builtin(__builtin_amdgcn_s_wait_asynccnt)
  __builtin_amdgcn_s_wait_asynccnt(0);
#else
  asm volatile("s_wait_asynccnt 0" ::: "memory");
#endif
#endif
  __syncthreads();

  const int tx = tid & (TW - 1);
  const int ty = tid >> 6;
  const float2 c = tile[ty + 1][tx + 1];

  float m0 = -3.402823466e38f, m1 = -3.402823466e38f;
#pragma unroll
  for (int dy = 0; dy < 3; ++dy)
#pragma unroll
    for (int dx = 0; dx < 3; ++dx) {
      const float2 n = tile[ty + dy][tx + dx];
      m0 = fmaxf(m0, n.x);
      m1 = fmaxf(m1, n.y);
    }

  const unsigned idx = (unsigned)(y0 + ty) * IMG_W + (unsigned)(x0 + tx);
  // channel 0 (objects)
  if (c.x == m0 && c.x > 0.5f) {
    const unsigned old = atomicAdd(&cnt[0], 1u);
    if (old < cap)
      cand[old] = ((u64)__float_as_uint(c.x) << 32) | (unsigned)(~idx);
  }
  // channel 1 (nose landmarks)
  if (c.y == m1 && c.y > 0.5f) {
    const unsigned old = atomicAdd(&cnt[1], 1u);
    if (old < cap)
      cand[(size_t)cap + old] = ((u64)__float_as_uint(c.y) << 32) | (unsigned)(~idx);
  }
}

// ---- Kernel 2: exact top-15 selection per channel ------------------------
__global__ __launch_bounds__(256)
void select_top_kernel(const u64* __restrict__ cand, const unsigned* __restrict__ cnt,
                       unsigned cap, u64* __restrict__ sel, unsigned* __restrict__ nsel) {
  const int ch = blockIdx.x;
  const u64* buf = cand + (size_t)ch * cap;
  unsigned n = cnt[ch];
  if (n > cap) n = cap;

  __shared__ u64 red[256];
  u64 prev = ~0ull;  // keys are never ~0 (value>0.5 => finite positive float)
  unsigned found = 0;

  for (int pass = 0; pass < N_OBJS; ++pass) {
    u64 best = 0;
    for (unsigned i = threadIdx.x; i < n; i += 256) {
      __builtin_prefetch(&buf[i + 2048], 0, 1);  // -> global_prefetch_b8
      const u64 k = buf[i];
      if (k < prev && k > best) best = k;
    }
    red[threadIdx.x] = best;
    __syncthreads();
#pragma unroll
    for (int s = 128; s > 0; s >>= 1) {
      if (threadIdx.x < s) {
        const u64 o = red[threadIdx.x + s];
        if (o > red[threadIdx.x]) red[threadIdx.x] = o;
      }
      __syncthreads();
    }
    const u64 winner = red[0];
    __syncthreads();
    if (winner == 0) break;  // no candidates left (uniform across block)
    if (threadIdx.x == 0) sel[ch * N_OBJS + pass] = winner;
    prev = winner;
    found = pass + 1;
  }
  if (threadIdx.x == 0) nsel[ch] = found;
}

// ---- Kernel 3: finalize (15,16) output -----------------------------------
__global__ void finalize_kernel(const float* __restrict__ szmap,
                                const float* __restrict__ offmap,
                                const u64* __restrict__ sel,
                                const unsigned* __restrict__ nsel,
                                float* __restrict__ out) {
  const int i = threadIdx.x;
  if (i >= N_OBJS) return;
  const float RATIO = 720.0f / 1080.0f;  // == 1280/1920 in f32
  float* row = out + i * 16;

  // boxes (cols 0..4)
  if ((unsigned)i < nsel[0]) {
    const u64 k = sel[i];
    const float score = __uint_as_float((unsigned)(k >> 32));
    const unsigned idx = ~(unsigned)(k & 0xffffffffu);
    const float cy = (float)(idx / IMG_W);
    const float cx = (float)(idx % IMG_W);
    const float s0 = szmap[(size_t)idx * 2 + 0];
    const float s1 = szmap[(size_t)idx * 2 + 1];
    row[0] = fmaxf(cy - s0 * 0.5f, 0.0f) * RATIO;
    row[1] = fmaxf(cx - s1 * 0.5f, 0.0f) * RATIO;
    row[2] = fminf(cy + s0 * 0.5f, (float)(IMG_H - 1)) * RATIO;
    row[3] = fminf(cx + s1 * 0.5f, (float)(IMG_W - 1)) * RATIO;
    row[4] = score;
  } else {
    row[0] = row[1] = row[2] = row[3] = row[4] = -1.0f;
  }

  // landmarks (cols 5..15)
  if ((unsigned)i < nsel[1]) {
    const u64 k = sel[N_OBJS + i];
    const float ns = __uint_as_float((unsigned)(k >> 32));
    const unsigned idx = ~(unsigned)(k & 0xffffffffu);
    const float py = (float)(idx / IMG_W) * RATIO;
    const float px = (float)(idx % IMG_W) * RATIO;
    const float* o = offmap + (size_t)idx * 8;
    float e[8];
#pragma unroll
    for (int j = 0; j < 4; ++j) {
      e[2 * j + 0] = py - o[2 * j + 0] * RATIO;
      e[2 * j + 1] = px - o[2 * j + 1] * RATIO;
    }
    row[5] = e[0]; row[6] = e[1]; row[7] = e[2]; row[8] = e[3];
    row[9] = py;   row[10] = px;
    row[11] = e[4]; row[12] = e[5]; row[13] = e[6]; row[14] = e[7];
    row[15] = ns;
  } else {
#pragma unroll
    for (int j = 5; j < 16; ++j) row[j] = -1.0f;
  }
}

// ---------------------------------------------------------------------------
extern "C" void kernel_launch(void* const* d_in, const int* in_sizes, int n_in,
                              void* d_out, int out_size, void* d_ws, size_t ws_size,
                              hipStream_t stream) {
  const float* hm  = (const float*)d_in[0];  // (1,1080,1920,2)
  const float* off = (const float*)d_in[1];  // (1,1080,1920,8)
  const float* sz  = (const float*)d_in[2];  // (1,1080,1920,2)
  float* out = (float*)d_out;                // 15*16 f32

  // Workspace: [0]=cnt[2] u32, [8]=nsel[2] u32, [16]=sel[2][15] u64 (ends 256),
  //            [256..] candidate buffers, cap entries per channel.
  unsigned* cnt  = (unsigned*)d_ws;
  unsigned* nsel = cnt + 2;
  u64* sel  = (u64*)((char*)d_ws + 16);
  u64* cand = (u64*)((char*)d_ws + 256);
  const size_t avail = ws_size > 256 ? ws_size - 256 : 0;
  size_t cap_sz = avail / (2 * sizeof(u64));  // per channel
  const size_t CAP_MAX = (size_t)1 << 19;     // 512K entries; expected ~230K
  if (cap_sz > CAP_MAX) cap_sz = CAP_MAX;
  const unsigned cap = (unsigned)cap_sz;

  init_ws_kernel<<<1, 32, 0, stream>>>(cnt);
  nms_collect_kernel<<<dim3(IMG_W / TW, IMG_H / TH), 256, 0, stream>>>(hm, cand, cnt, cap);
  select_top_kernel<<<2, 256, 0, stream>>>(cand, cnt, cap, sel, nsel);
  finalize_kernel<<<1, 32, 0, stream>>>(sz, off, sel, nsel, out);
}